// KabschDecoder_35338990911881
// MI455X (gfx1250) — compile-verified
//
#include <hip/hip_runtime.h>
#include <hip/hip_bf16.h>

typedef float v2f __attribute__((ext_vector_type(2)));
typedef float v8f __attribute__((ext_vector_type(8)));

namespace {
constexpr int   kB            = 4;
constexpr int   kS            = 64;
constexpr int   kN            = 65536;
constexpr float kInv2Pi       = 0.15915494309189535f;        // v_sin/v_cos take revolutions
constexpr float kK2           = 7.0f * 1.4426950408889634f;  // SIGMOID_SLOPE * log2(e)
constexpr int   kThreads      = 256;                          // 8 waves (wave32)
constexpr int   kWavesPerBlk  = kThreads / 32;
constexpr int   kTilesPerWave = 4;                            // 4 consecutive 16-point tiles
constexpr int   kNTiles       = kN / 16;                      // 4096
constexpr int   kGroupsPerBS  = kNTiles / kTilesPerWave;      // 1024
constexpr int   kTotalWaves   = kB * (kS / 16) * kGroupsPerBS;// 16384
constexpr int   kBlocks       = kTotalWaves / kWavesPerBlk;   // 2048
}

__global__ __launch_bounds__(kThreads)
void box_point_weights_wmma(const float* __restrict__ pos,    // (B,S,3)
                            const float* __restrict__ dims,   // (B,S,3)
                            const float* __restrict__ rot,    // (B,S,1)
                            const float* __restrict__ pts,    // (B,N,3)
                            const unsigned char* __restrict__ vmask, // (B,N) bool
                            float* __restrict__ out)          // (B,S,N)
{
  const int lane = threadIdx.x & 31;
  const int lo   = lane & 15;          // point / box sub-index
  const int hi   = lane >> 4;          // 0: K=0,1 half   1: K=2,3 half

  const int gw  = blockIdx.x * kWavesPerBlk + (threadIdx.x >> 5);
  const int bs  = gw >> 10;            // (b, s_tile) pair, 0..15
  const int g   = gw & 1023;           // wave-group within (b, s_tile)
  const int b   = bs >> 2;
  const int st  = bs & 3;
  const int sbase = st * 16;
  const int nt0   = g * kTilesPerWave;

  // ---------------- per-wave A-operand setup (box m = sbase + lo) --------
  const int   m      = sbase + lo;
  const int   boxIdx = b * kS + m;
  const float theta  = rot[boxIdx];
  const float tx     = pos[boxIdx * 3 + 0];
  const float ty     = pos[boxIdx * 3 + 1];
  const float tz     = pos[boxIdx * 3 + 2];
  const float hx2    = kK2 * 0.5f * dims[boxIdx * 3 + 0];   // K2 * half-dim
  const float hy2    = kK2 * 0.5f * dims[boxIdx * 3 + 1];
  const float hz2    = kK2 * 0.5f * dims[boxIdx * 3 + 2];

  const float tn = theta * kInv2Pi;
  const float c  = __builtin_amdgcn_cosf(tn);
  const float s  = __builtin_amdgcn_sinf(tn);
  // inverse transform translation column
  const float tix = -(c * tx + s * ty);
  const float tiy =  (s * tx - c * ty);
  const float tiz = -tz;

  // A rows of box_T_sensor: x:[c,s,0,tix]  y:[-s,c,0,tiy]  z:[0,0,1,tiz]
  const v2f aX = hi ? (v2f){0.0f, tix} : (v2f){ c,   s};
  const v2f aY = hi ? (v2f){0.0f, tiy} : (v2f){-s,   c};
  const v2f aZ = hi ? (v2f){1.0f, tiz} : (v2f){0.0f, 0.0f};

  const v8f zero = {};

  // Broadcast K2*halfdim into C/D layout: A=[0,0,0,K2h] x B with row K=3 == 1
  const v2f aHx = hi ? (v2f){0.0f, hx2} : (v2f){0.0f, 0.0f};
  const v2f aHy = hi ? (v2f){0.0f, hy2} : (v2f){0.0f, 0.0f};
  const v2f aHz = hi ? (v2f){0.0f, hz2} : (v2f){0.0f, 0.0f};
  const v2f bOne = hi ? (v2f){0.0f, 1.0f} : (v2f){0.0f, 0.0f};

  const v8f HX = __builtin_amdgcn_wmma_f32_16x16x4_f32(false, aHx, false, bOne,
                                                       (short)0, zero, false, false);
  const v8f HY = __builtin_amdgcn_wmma_f32_16x16x4_f32(false, aHy, false, bOne,
                                                       (short)0, zero, false, false);
  const v8f HZ = __builtin_amdgcn_wmma_f32_16x16x4_f32(false, aHz, false, bOne,
                                                       (short)0, zero, false, false);

  // ---------------- per-tile loop ----------------------------------------
  #pragma unroll
  for (int t = 0; t < kTilesPerWave; ++t) {
    const int nbase = (nt0 + t) * 16;
    const int n     = nbase + lo;

    const float* p = pts + ((size_t)b * kN + n) * 3;
    // prefetch next tile's points (speculative; OOB silently dropped)
    __builtin_prefetch(p + 48, 0, 0);

    const float msk = vmask[(size_t)b * kN + n] ? 1.0f : 0.0f;
    const float px = p[0] * msk;
    const float py = p[1] * msk;
    const float pz = p[2] * msk;

    // B = homogeneous points, 4x16: lanes 0-15 hold K=0,1; lanes 16-31 K=2,3
    const v2f bv = { hi ? pz : px, hi ? 1.0f : py };

    const v8f X = __builtin_amdgcn_wmma_f32_16x16x4_f32(false, aX, false, bv,
                                                        (short)0, zero, false, false);
    const v8f Y = __builtin_amdgcn_wmma_f32_16x16x4_f32(false, aY, false, bv,
                                                        (short)0, zero, false, false);
    const v8f Z = __builtin_amdgcn_wmma_f32_16x16x4_f32(false, aZ, false, bv,
                                                        (short)0, zero, false, false);

    // lane owns (s = sbase + r + 8*hi, n); row stride folds into IOFFSET
    float* op = out + ((size_t)(b * kS + sbase + 8 * hi)) * kN + n;

    #pragma unroll
    for (int r = 0; r < 8; ++r) {
      // sig(a)sig(b)sig(c) = 1 / ((1+e^-a)(1+e^-b)(1+e^-c)),
      // -a = 7(|x'|-hx) -> base-2: K2*|x'| - K2*hx
      const float ex = __builtin_amdgcn_exp2f(__builtin_fmaf(kK2, __builtin_fabsf(X[r]), -HX[r]));
      const float ey = __builtin_amdgcn_exp2f(__builtin_fmaf(kK2, __builtin_fabsf(Y[r]), -HY[r]));
      const float ez = __builtin_amdgcn_exp2f(__builtin_fmaf(kK2, __builtin_fabsf(Z[r]), -HZ[r]));
      const float den = (1.0f + ex) * (1.0f + ey) * (1.0f + ez);
      const float w   = __builtin_amdgcn_rcpf(den);
      __builtin_nontemporal_store(w, op + (size_t)r * kN);
    }
  }
}

extern "C" void kernel_launch(void* const* d_in, const int* in_sizes, int n_in,
                              void* d_out, int out_size, void* d_ws, size_t ws_size,
                              hipStream_t stream) {
  const float*         pos   = (const float*)d_in[0];
  const float*         dims  = (const float*)d_in[1];
  const float*         rot   = (const float*)d_in[2];
  const float*         pts   = (const float*)d_in[3];
  const unsigned char* vmask = (const unsigned char*)d_in[4];
  float*               out   = (float*)d_out;

  box_point_weights_wmma<<<kBlocks, kThreads, 0, stream>>>(pos, dims, rot, pts, vmask, out);
}